// TVLoss_63634235457665
// MI455X (gfx1250) — compile-verified
//
#include <hip/hip_runtime.h>
#include <hip/hip_bf16.h>

#ifndef __has_builtin
#define __has_builtin(x) 0
#endif

#if __has_builtin(__builtin_amdgcn_global_load_async_to_lds_b128)
#define HAVE_ASYNC_LDS 1
#else
#define HAVE_ASYNC_LDS 0
#endif

#define B_DIM 64
#define L_DIM 8192
#define C_DIM 64
#define ALPHA 0.1f

#define TOKS 128           // tokens per block in kernel 1
#define K1_THREADS 128
#define PAD 68             // floats per token in LDS (272B: 16B-aligned, <=2-way bank conflict)

#if HAVE_ASYNC_LDS
typedef int v4i __attribute__((ext_vector_type(4)));
typedef __attribute__((address_space(1))) v4i* global_v4i_ptr;
typedef __attribute__((address_space(3))) v4i* shared_v4i_ptr;
#endif

__device__ __forceinline__ unsigned ballot32(bool p) {
#if __has_builtin(__builtin_amdgcn_ballot_w32)
    return __builtin_amdgcn_ballot_w32(p);
#else
    return (unsigned)__ballot(p);
#endif
}

// ---------------- Kernel 1: per-token argmax over C=64 ----------------
__global__ __launch_bounds__(K1_THREADS) void k_argmax(
    const float* __restrict__ y_pred, unsigned char* __restrict__ labels) {
    __shared__ float tile[TOKS * PAD];

    const int tid = threadIdx.x;
    const long long gbase = (long long)blockIdx.x * (TOKS * C_DIM);
    const float* src = y_pred + gbase;

    // Stage 128 tokens x 64 channels (8192 floats) into LDS, padded layout.
    #pragma unroll
    for (int it = 0; it < 16; ++it) {
        const int q = tid + it * K1_THREADS;  // float4 index 0..2047
        const int f = q << 2;                 // float index
        const int tok = f >> 6;
        const int c = f & 63;
        float* dst = &tile[tok * PAD + c];    // byte offset multiple of 16
#if HAVE_ASYNC_LDS
        __builtin_amdgcn_global_load_async_to_lds_b128(
            (global_v4i_ptr)(src + f),
            (shared_v4i_ptr)dst,
            0, 0);
#else
        const float4 v = *(const float4*)(src + f);
        *(float4*)dst = v;
#endif
    }
#if HAVE_ASYNC_LDS
#if __has_builtin(__builtin_amdgcn_s_wait_asynccnt)
    __builtin_amdgcn_s_wait_asynccnt(0);
#else
    asm volatile("s_wait_asynccnt 0" ::: "memory");
#endif
#endif
    __syncthreads();

    // Thread t argmaxes token t (first-max-wins, matches jnp.argmax).
    const int base = tid * PAD;
    float best = tile[base];
    int bidx = 0;
    #pragma unroll
    for (int c = 1; c < C_DIM; ++c) {
        const float v = tile[base + c];
        if (v > best) { best = v; bidx = c; }
    }
    labels[(long long)blockIdx.x * TOKS + tid] = (unsigned char)bidx;
}

// ---- Kernel 2: per-row TV over valid-compacted labels (segmented scan) ----
__global__ __launch_bounds__(1024) void k_rowscan(
    const int* __restrict__ y_true, const unsigned char* __restrict__ labels,
    float* __restrict__ row_loss, int* __restrict__ row_cnt) {
    const int row  = blockIdx.x;
    const int lane = threadIdx.x & 31;
    const int wave = threadIdx.x >> 5;        // 0..31 segments of 256
    const long long rb = (long long)row * L_DIM;
    const int segBase = wave * 256;

    __shared__ int s_sum[32], s_cnt[32], s_first[32], s_last[32], s_has[32];

    int isum = 0, vcnt = 0;
    int carry_lab = 0;  bool carry_valid = false;   // last valid label so far in segment
    int seg_first = 0;  bool seg_has = false;       // first valid label of segment

    #pragma unroll
    for (int k = 0; k < 8; ++k) {
        const long long idx = rb + segBase + k * 32 + lane;
        const int  lab = (int)labels[idx];
        const bool v   = (y_true[idx] != -1);

        const unsigned mask  = ballot32(v);
        const unsigned below = mask & ((1u << lane) - 1u);
        const int pl      = below ? (31 - __builtin_clz(below)) : 0;
        const int nb_lab  = __shfl(lab, pl, 32);          // all lanes execute
        const int  prev   = below ? nb_lab : carry_lab;
        const bool hasp   = below ? true   : carry_valid;
        if (v && hasp) { const int d = lab - prev; isum += (d < 0) ? -d : d; }
        vcnt += v ? 1 : 0;

        const int msb = mask ? (31 - __builtin_clz(mask)) : 0;
        const int lsb = mask ? __builtin_ctz(mask) : 0;
        const int last_lab  = __shfl(lab, msb, 32);
        const int first_lab = __shfl(lab, lsb, 32);
        if (mask) {
            if (!seg_has) { seg_first = first_lab; seg_has = true; }
            carry_lab = last_lab; carry_valid = true;
        }
    }
    // wave-reduce the per-lane partials
    #pragma unroll
    for (int off = 16; off > 0; off >>= 1) {
        isum += __shfl_xor(isum, off, 32);
        vcnt += __shfl_xor(vcnt, off, 32);
    }
    if (lane == 0) {
        s_sum[wave] = isum;  s_cnt[wave] = vcnt;
        s_first[wave] = seg_first;  s_last[wave] = carry_lab;
        s_has[wave] = seg_has ? 1 : 0;
    }
    __syncthreads();

    // Wave 0 stitches the 32 segments (cross-segment |first - prev_last| terms).
    if (threadIdx.x < 32) {
        const int w = lane;
        int ssum = s_sum[w];
        int scnt = s_cnt[w];
        const int  sfir = s_first[w];
        const int  slas = s_last[w];
        const bool shas = (s_has[w] != 0);

        const unsigned m2     = ballot32(shas);
        const unsigned below2 = m2 & ((1u << w) - 1u);
        const int pl2      = below2 ? (31 - __builtin_clz(below2)) : 0;
        const int prevlast = __shfl(slas, pl2, 32);
        if (shas && below2) { const int d = sfir - prevlast; ssum += (d < 0) ? -d : d; }

        #pragma unroll
        for (int off = 16; off > 0; off >>= 1) {
            ssum += __shfl_xor(ssum, off, 32);
            scnt += __shfl_xor(scnt, off, 32);
        }
        if (w == 0) {
            // xgrad = 0.5 * ssum (exact); gradient(): 0 if xgrad==0 else xgrad-0.5
            row_loss[row] = (ssum == 0) ? 0.0f : (0.5f * (float)ssum - 0.5f);
            row_cnt[row]  = scnt;
        }
    }
}

// ---------------- Kernel 3: final scalar ----------------
__global__ __launch_bounds__(32) void k_final(
    const float* __restrict__ row_loss, const int* __restrict__ row_cnt,
    float* __restrict__ out) {
    const int lane = threadIdx.x;
    float s = row_loss[lane] + row_loss[lane + 32];
    int   c = row_cnt[lane]  + row_cnt[lane + 32];
    #pragma unroll
    for (int off = 16; off > 0; off >>= 1) {
        s += __shfl_xor(s, off, 32);
        c += __shfl_xor(c, off, 32);
    }
    if (lane == 0) out[0] = ALPHA * s / (float)c;
}

extern "C" void kernel_launch(void* const* d_in, const int* in_sizes, int n_in,
                              void* d_out, int out_size, void* d_ws, size_t ws_size,
                              hipStream_t stream) {
    const int*   y_true = (const int*)d_in[0];     // [B, L] int32
    const float* y_pred = (const float*)d_in[1];   // [B, L, C] f32
    float* out = (float*)d_out;

    // workspace layout: u8 labels [B*L] (512 KiB), then f32 row_loss[64], i32 row_cnt[64]
    unsigned char* labels = (unsigned char*)d_ws;
    char* base = (char*)d_ws;
    float* row_loss = (float*)(base + (size_t)B_DIM * L_DIM);
    int*   row_cnt  = (int*)(base + (size_t)B_DIM * L_DIM + B_DIM * sizeof(float));

    const int n_tokens = B_DIM * L_DIM;            // 524288
    k_argmax <<<n_tokens / TOKS, K1_THREADS, 0, stream>>>(y_pred, labels);
    k_rowscan<<<B_DIM, 1024, 0, stream>>>(y_true, labels, row_loss, row_cnt);
    k_final  <<<1, 32, 0, stream>>>(row_loss, row_cnt, out);
}